// MOElayer_63488206569763
// MI455X (gfx1250) — compile-verified
//
#include <hip/hip_runtime.h>

typedef __attribute__((ext_vector_type(2))) float v2f;
typedef __attribute__((ext_vector_type(8))) float v8f;

#define NTOK   131072
#define DIN    128
#define NOE    8
#define TPB    128    // tokens per block
#define LSTR   132    // padded LDS row stride (mod 64 = 4 -> conflict-free b64 loads)

__launch_bounds__(256)
__global__ void moe_fused_kernel(const float* __restrict__ x,
                                 const float* __restrict__ router_w,
                                 const float* __restrict__ expert_w,
                                 const float* __restrict__ expert_b,
                                 float* __restrict__ out)
{
    __shared__ float xs[TPB * LSTR];    // ~67.6 KB: x tile [token][d], padded rows
    __shared__ float wsh[DIN * LSTR];   // ~67.6 KB: current expert W [o][d], padded
    __shared__ float cws[TPB * NOE];    // 4 KB: combine weights [token][e]

    const int tid  = threadIdx.x;
    const int lane = tid & 31;
    const int wave = tid >> 5;
    const int half = lane >> 4;         // 0 or 1 (wave32 halves)
    const int l16  = lane & 15;
    const int tok0 = blockIdx.x * TPB;  // block's first token
    const int tw   = wave * 16;         // wave's token offset within tile

    // ---- stage x tile: coalesced float4 global reads -> padded LDS rows ----
    {
        const float4* xg = (const float4*)(x + (size_t)tok0 * DIN);
        for (int i = tid; i < TPB * (DIN / 4); i += 256) {
            const int row = i >> 5;       // DIN/4 = 32 float4 per row
            const int c4  = i & 31;
            *(float4*)(xs + row * LSTR + c4 * 4) = xg[i];
        }
    }
    __syncthreads();

    // ---- fused router: threads 0..127, one token each, x from LDS ----
    if (tid < TPB) {
        float logit[NOE];
        #pragma unroll
        for (int e = 0; e < NOE; ++e) logit[e] = 0.0f;
        const float* xrow = xs + tid * LSTR;
        for (int d = 0; d < DIN; d += 4) {
            float4 xv = *(const float4*)(xrow + d);
            #pragma unroll
            for (int e = 0; e < NOE; ++e) {
                float4 wv = *(const float4*)(router_w + e * DIN + d);
                logit[e] = fmaf(xv.x, wv.x, logit[e]);
                logit[e] = fmaf(xv.y, wv.y, logit[e]);
                logit[e] = fmaf(xv.z, wv.z, logit[e]);
                logit[e] = fmaf(xv.w, wv.w, logit[e]);
            }
        }
        // top-2 (strict > matches lax.top_k lowest-index tie-break)
        int e0 = 0; float l0 = logit[0];
        #pragma unroll
        for (int e = 1; e < NOE; ++e) if (logit[e] > l0) { l0 = logit[e]; e0 = e; }
        int e1 = (e0 == 0) ? 1 : 0; float l1 = logit[e1];
        #pragma unroll
        for (int e = 0; e < NOE; ++e)
            if (e != e0 && logit[e] > l1) { l1 = logit[e]; e1 = e; }
        float w0 = 1.0f / (1.0f + __expf(l1 - l0));  // softmax over {l0,l1}
        float w1 = 1.0f - w0;
        #pragma unroll
        for (int e = 0; e < NOE; ++e) cws[tid * NOE + e] = 0.0f;
        cws[tid * NOE + e0] = w0;
        cws[tid * NOE + e1] = w1;
    }

    // accumulators: 8 output-dim blocks x (16 tok x 16 out) tile each
    v8f acc[8];
    #pragma unroll
    for (int ob = 0; ob < 8; ++ob) acc[ob] = v8f{};

    for (int e = 0; e < NOE; ++e) {
        __syncthreads();  // previous wsh use done (also covers router barrier)
        {   // stage W_e into padded LDS
            const float4* wg = (const float4*)(expert_w + (size_t)e * DIN * DIN);
            for (int i = tid; i < DIN * (DIN / 4); i += 256) {
                const int row = i >> 5;
                const int c4  = i & 31;
                *(float4*)(wsh + row * LSTR + c4 * 4) = wg[i];
            }
        }
        __syncthreads();

        // per-lane combine weights for the 8 token rows this lane's D VGPRs cover
        // (same address across each half-wave -> LDS broadcast, no conflict)
        float cwr[8];
        #pragma unroll
        for (int v = 0; v < 8; ++v)
            cwr[v] = cws[(tw + v + 8 * half) * NOE + e];

        #pragma unroll
        for (int ob = 0; ob < 8; ++ob) {
            v8f y = v8f{};
            const float bvec = expert_b[e * DIN + ob * 16 + l16];  // o fixed per lane
            // A: x[token=tw+l16][k + 2*half + {0,1}]  (f32 16x4 layout)
            const float* abase = xs  + (tw + l16) * LSTR + half * 2;
            // B: W_e[o=ob*16+l16][k + 2*half + {0,1}] (4x16, contraction over d)
            const float* bbase = wsh + (ob * 16 + l16) * LSTR + half * 2;
            #pragma unroll
            for (int ks = 0; ks < DIN; ks += 4) {
                v2f a = *(const v2f*)(abase + ks);
                v2f b = *(const v2f*)(bbase + ks);
                y = __builtin_amdgcn_wmma_f32_16x16x4_f32(
                        false, a, false, b, (short)0, y, false, false);
            }
            // out += cw[tok,e] * (y + b_e[o]); tok varies per VGPR row, o per lane
            #pragma unroll
            for (int v = 0; v < 8; ++v)
                acc[ob][v] = fmaf(cwr[v], y[v] + bvec, acc[ob][v]);
        }
    }

    // ---- store: token = tok0+tw+v+8*half, o = ob*16+l16 (64B coalesced runs) ----
    #pragma unroll
    for (int ob = 0; ob < 8; ++ob) {
        #pragma unroll
        for (int v = 0; v < 8; ++v) {
            const int token = tok0 + tw + v + 8 * half;
            out[(size_t)token * DIN + ob * 16 + l16] = acc[ob][v];
        }
    }
}

extern "C" void kernel_launch(void* const* d_in, const int* in_sizes, int n_in,
                              void* d_out, int out_size, void* d_ws, size_t ws_size,
                              hipStream_t stream) {
    (void)in_sizes; (void)n_in; (void)d_ws; (void)ws_size; (void)out_size;
    const float* x        = (const float*)d_in[0];
    const float* router_w = (const float*)d_in[1];
    const float* expert_w = (const float*)d_in[2];
    const float* expert_b = (const float*)d_in[3];
    // d_in[4] is topk (==2); the top-2 path is specialized in-kernel.
    float* out = (float*)d_out;

    dim3 grid(NTOK / TPB);   // 1024 blocks
    dim3 block(256);         // 8 wave32s
    moe_fused_kernel<<<grid, block, 0, stream>>>(x, router_w, expert_w, expert_b, out);
}